// DNDLSTM_27015344291895
// MI455X (gfx1250) — compile-verified
//
#include <hip/hip_runtime.h>
#include <math.h>

// ---------------- problem constants ----------------
#define IN_DIM   128
#define HID      128
#define OUT_DIM  16
#define DICT_LEN 400000
#define PRE_DIM  640      // (N_GATES+1)*HID

// ---------------- launch geometry ------------------
#define NB1 1024   // blocks for keys pass
#define NB3 1024   // blocks for softmax-exp pass
#define NB5 1024   // blocks for vals pass
#define NBP 40     // blocks for preact WMMA kernel (one 16-row tile each)

// ---------------- workspace layout (float offsets) --
#define WS_MAXSIM 0
#define WS_SUMW   1
#define WS_D2     16                       // 400000 floats; reused in-place for exp weights
#define WS_BMIN   (WS_D2 + DICT_LEN)       // NB1 floats
#define WS_BSUM   (WS_BMIN + NB1)          // NB3 floats
#define WS_VPART  (WS_BSUM + NB3)          // NB5*128 floats
#define WS_PRE    (WS_VPART + NB5 * 128)   // PRE_DIM floats
// total ~534K floats (~2.14 MB)

typedef float v2f __attribute__((ext_vector_type(2)));
typedef float v8f __attribute__((ext_vector_type(8)));

// ---------------------------------------------------------------------------
// K0: preactivation GEMV via V_WMMA_F32_16X16X4_F32 (fp32-exact).
//   pre[n] = sum_k x[k]*W_i2h[n,k] + h[k]*W_h2h[n,k] + b_i2h[n] + b_h2h[n]
// One block per 16-output tile (40 blocks). The block stages its two 16x128
// weight tiles in LDS via coalesced float4 loads, then the 128-long K dim is
// split across 8 waves (16 k each -> 4+4 chained WMMAs per wave, B fragments
// from LDS, no per-WMMA global-load stalls). Wave partials combine in LDS.
//
// Fragment layouts (16x16x4 f32):
//   A 16x4 : lanes 0-15 hold K=0,1 (vgpr0,1); lanes 16-31 hold K=2,3.
//            Only row M=0 carries the vector. Lane 0 needs vec[kb..kb+1],
//            lane 16 needs vec[kb+2..kb+3] == vec[kb+khalf .. +1] for both;
//            built branchlessly: unconditional 8B LDS load * lane mask.
//   B 4x16 : vgpr0 lanes0-15 -> K=0,N=lane; lanes16-31 -> K=2. vgpr1 -> K=1/K=3.
//   D      : row M=0 result lands in acc[0] of lanes 0..15 (N = lane).
// ---------------------------------------------------------------------------
__global__ void k0_preact(const float* __restrict__ x, const float* __restrict__ h,
                          const float* __restrict__ W_i2h, const float* __restrict__ b_i2h,
                          const float* __restrict__ W_h2h, const float* __restrict__ b_h2h,
                          float* __restrict__ ws) {
  __shared__ float wt_x[16 * 128];   // W_i2h tile rows nb..nb+15
  __shared__ float wt_h[16 * 128];   // W_h2h tile rows nb..nb+15
  __shared__ float xs[IN_DIM], hs[HID];
  __shared__ float part[8 * 16];

  const int t    = threadIdx.x;      // 256 threads = 8 waves
  const int lane = t & 31;
  const int wave = t >> 5;
  const int nb   = blockIdx.x * 16;

  // cooperative staging: 2048 floats per tile = 512 float4, 256 threads x 2
  {
    const float4* src_x = (const float4*)(W_i2h + (size_t)nb * IN_DIM);
    const float4* src_h = (const float4*)(W_h2h + (size_t)nb * HID);
    float4* dst_x = (float4*)wt_x;
    float4* dst_h = (float4*)wt_h;
    dst_x[t]        = src_x[t];
    dst_x[t + 256]  = src_x[t + 256];
    dst_h[t]        = src_h[t];
    dst_h[t + 256]  = src_h[t + 256];
  }
  if (t < IN_DIM) { xs[t] = x[t]; hs[t] = h[t]; }
  __syncthreads();

  const int   n     = lane & 15;
  const int   khalf = (lane >> 4) << 1;           // 0 for lanes 0-15, 2 for 16-31
  const float amask = (n == 0) ? 1.0f : 0.0f;     // only lanes 0 and 16 feed A row 0
  const int   k0    = wave * 16;                  // this wave's 16-wide K slice

  v8f acc = {};
  #pragma unroll
  for (int kk = 0; kk < 16; kk += 4) {            // x * W_i2h part
    const int kb = k0 + kk;
    v2f av = *(const v2f*)&xs[kb + khalf];        // unconditional 8B LDS load
    v2f a;  a.x = av.x * amask;  a.y = av.y * amask;
    v2f b  = *(const v2f*)&wt_x[n * 128 + kb + khalf];
    acc = __builtin_amdgcn_wmma_f32_16x16x4_f32(false, a, false, b,
                                                (short)0, acc, false, false);
  }
  #pragma unroll
  for (int kk = 0; kk < 16; kk += 4) {            // h * W_h2h part
    const int kb = k0 + kk;
    v2f av = *(const v2f*)&hs[kb + khalf];
    v2f a;  a.x = av.x * amask;  a.y = av.y * amask;
    v2f b  = *(const v2f*)&wt_h[n * 128 + kb + khalf];
    acc = __builtin_amdgcn_wmma_f32_16x16x4_f32(false, a, false, b,
                                                (short)0, acc, false, false);
  }
  if (lane < 16) part[wave * 16 + lane] = acc[0];
  __syncthreads();

  if (t < 16) {
    float s = 0.f;
    #pragma unroll
    for (int w = 0; w < 8; ++w) s += part[w * 16 + t];
    ws[WS_PRE + nb + t] = s + b_i2h[nb + t] + b_h2h[nb + t];
  }
}

// ---------------------------------------------------------------------------
// K1: stream mem_keys once. Per wave: one row/iter, each lane holds 4 elems.
// Computes d2[row], block-min of d2 (for softmax max), and writes the shifted
// keys_new[row+1] from the same data (dword stores: dest is 4B-aligned only).
// ---------------------------------------------------------------------------
__global__ void k1_keys_pass(const float* __restrict__ x,
                             const float* __restrict__ mem_keys,
                             float* __restrict__ keys_new,
                             float* __restrict__ ws) {
  const int lane = threadIdx.x & 31;
  const int wave = threadIdx.x >> 5;
  const int wpb  = blockDim.x >> 5;
  const int gw   = blockIdx.x * wpb + wave;
  const int nW   = gridDim.x * wpb;

  float4 x4 = ((const float4*)x)[lane];
  float wmin = 3.0e38f;

  for (int row = gw; row < DICT_LEN; row += nW) {
    float4 k4 = ((const float4*)(mem_keys + (size_t)row * IN_DIM))[lane];
    float dx = k4.x - x4.x, dy = k4.y - x4.y;
    float dz = k4.z - x4.z, dw = k4.w - x4.w;
    float d = dx*dx + dy*dy + dz*dz + dw*dw;
    #pragma unroll
    for (int m = 16; m >= 1; m >>= 1) d += __shfl_xor(d, m, 32);
    if (lane == 0) {
      ws[WS_D2 + row] = d;
      wmin = fminf(wmin, d);
    }
    if (row < DICT_LEN - 1) {
      float* dst = keys_new + (size_t)(row + 1) * IN_DIM + lane * 4;
      dst[0] = k4.x; dst[1] = k4.y; dst[2] = k4.z; dst[3] = k4.w;
    }
  }
  __shared__ float smin[32];
  if (lane == 0) smin[wave] = wmin;
  __syncthreads();
  if (threadIdx.x == 0) {
    float m = smin[0];
    for (int i = 1; i < wpb; ++i) m = fminf(m, smin[i]);
    ws[WS_BMIN + blockIdx.x] = m;
  }
}

// K2: reduce block mins -> maxsim = -sqrt(min_d2 + 1e-6)
__global__ void k2_reduce_min(float* __restrict__ ws) {
  __shared__ float s[256];
  float m = 3.0e38f;
  for (int i = threadIdx.x; i < NB1; i += blockDim.x) m = fminf(m, ws[WS_BMIN + i]);
  s[threadIdx.x] = m;
  __syncthreads();
  for (int off = blockDim.x >> 1; off > 0; off >>= 1) {
    if (threadIdx.x < off) s[threadIdx.x] = fminf(s[threadIdx.x], s[threadIdx.x + off]);
    __syncthreads();
  }
  if (threadIdx.x == 0) ws[WS_MAXSIM] = -sqrtf(s[0] + 1e-6f);
}

// K3: w[i] = exp(-sqrt(d2+eps) - maxsim), stored in place over d2; block partial sums.
__global__ void k3_exp_pass(float* __restrict__ ws) {
  const float maxsim = ws[WS_MAXSIM];
  float acc = 0.f;
  for (int i = blockIdx.x * blockDim.x + threadIdx.x; i < DICT_LEN;
       i += gridDim.x * blockDim.x) {
    float sim = -sqrtf(ws[WS_D2 + i] + 1e-6f);
    float w = expf(sim - maxsim);
    ws[WS_D2 + i] = w;
    acc += w;
  }
  __shared__ float s[256];
  s[threadIdx.x] = acc;
  __syncthreads();
  for (int off = blockDim.x >> 1; off > 0; off >>= 1) {
    if (threadIdx.x < off) s[threadIdx.x] += s[threadIdx.x + off];
    __syncthreads();
  }
  if (threadIdx.x == 0) ws[WS_BSUM + blockIdx.x] = s[0];
}

// K4: reduce partial sums -> sumw
__global__ void k4_reduce_sum(float* __restrict__ ws) {
  __shared__ float s[256];
  float a = 0.f;
  for (int i = threadIdx.x; i < NB3; i += blockDim.x) a += ws[WS_BSUM + i];
  s[threadIdx.x] = a;
  __syncthreads();
  for (int off = blockDim.x >> 1; off > 0; off >>= 1) {
    if (threadIdx.x < off) s[threadIdx.x] += s[threadIdx.x + off];
    __syncthreads();
  }
  if (threadIdx.x == 0) ws[WS_SUMW] = s[0];
}

// ---------------------------------------------------------------------------
// K5: stream mem_vals once. Weighted accumulation of sum_row w[row]*vals[row,:]
// into per-block 128-float partials + shifted vals_new[row+1] from same data.
// ---------------------------------------------------------------------------
__global__ void k5_vals_pass(const float* __restrict__ mem_vals,
                             float* __restrict__ vals_new,
                             float* __restrict__ ws) {
  const int lane = threadIdx.x & 31;
  const int wave = threadIdx.x >> 5;
  const int wpb  = blockDim.x >> 5;
  const int gw   = blockIdx.x * wpb + wave;
  const int nW   = gridDim.x * wpb;

  float4 acc = {0.f, 0.f, 0.f, 0.f};
  for (int row = gw; row < DICT_LEN; row += nW) {
    float4 v4 = ((const float4*)(mem_vals + (size_t)row * HID))[lane];
    float w = ws[WS_D2 + row];
    acc.x += w * v4.x; acc.y += w * v4.y;
    acc.z += w * v4.z; acc.w += w * v4.w;
    if (row < DICT_LEN - 1) {
      float* dst = vals_new + (size_t)(row + 1) * HID + lane * 4;
      dst[0] = v4.x; dst[1] = v4.y; dst[2] = v4.z; dst[3] = v4.w;
    }
  }
  __shared__ float sacc[8 * 128];
  float* myrow = sacc + wave * 128 + lane * 4;
  myrow[0] = acc.x; myrow[1] = acc.y; myrow[2] = acc.z; myrow[3] = acc.w;
  __syncthreads();
  if (threadIdx.x < 128) {
    float s = 0.f;
    for (int wv = 0; wv < 8; ++wv) s += sacc[wv * 128 + threadIdx.x];
    ws[WS_VPART + blockIdx.x * 128 + threadIdx.x] = s;
  }
}

// ---------------------------------------------------------------------------
// K6: scalar finalize (one block). m_t, gates, c_t/h_t, actor softmax, critic.
// ---------------------------------------------------------------------------
__global__ void k6_finalize(const float* __restrict__ x, const float* __restrict__ c,
                            const float* __restrict__ W_actor, const float* __restrict__ b_actor,
                            const float* __restrict__ W_critic, const float* __restrict__ b_critic,
                            float* __restrict__ out,
                            float* __restrict__ keys_new, float* __restrict__ vals_new,
                            float* __restrict__ ws) {
  __shared__ float hts[HID], lg[OUT_DIM];
  const int t = threadIdx.x;
  const float sumw = ws[WS_SUMW];

  if (t < HID) {
    float mraw = 0.f;
    for (int b = 0; b < NB5; ++b) mraw += ws[WS_VPART + b * 128 + t];
    float mt = tanhf(mraw / sumw);

    float f_ = 1.f / (1.f + expf(-ws[WS_PRE + t]));
    float i_ = 1.f / (1.f + expf(-ws[WS_PRE + HID + t]));
    float o_ = 1.f / (1.f + expf(-ws[WS_PRE + 2 * HID + t]));
    float r_ = 1.f / (1.f + expf(-ws[WS_PRE + 3 * HID + t]));
    float cn = tanhf(ws[WS_PRE + 4 * HID + t]);
    float ct = f_ * c[t] + i_ * cn + r_ * mt;
    float ht = o_ * tanhf(ct);
    hts[t] = ht;
    out[17 + t]       = ht;   // h_t at out[17..144]
    out[17 + HID + t] = ct;   // c_t at out[145..272]
    keys_new[t] = x[t];       // keys_new row 0 = x
    vals_new[t] = ct;         // vals_new row 0 = c_t
  }
  __syncthreads();

  if (t < OUT_DIM) {
    float a = b_actor[t];
    for (int k = 0; k < HID; ++k) a += hts[k] * W_actor[t * HID + k];
    lg[t] = a;
  } else if (t == OUT_DIM) {
    float a = b_critic[0];
    for (int k = 0; k < HID; ++k) a += hts[k] * W_critic[k];
    out[16] = a;              // v
  }
  __syncthreads();
  if (t == 0) {
    float mx = lg[0];
    for (int i = 1; i < OUT_DIM; ++i) mx = fmaxf(mx, lg[i]);
    float se = 0.f, e[OUT_DIM];
    for (int i = 0; i < OUT_DIM; ++i) { e[i] = expf(lg[i] - mx); se += e[i]; }
    for (int i = 0; i < OUT_DIM; ++i) out[i] = e[i] / se;   // pi
  }
}

extern "C" void kernel_launch(void* const* d_in, const int* in_sizes, int n_in,
                              void* d_out, int out_size, void* d_ws, size_t ws_size,
                              hipStream_t stream) {
  const float* x        = (const float*)d_in[0];
  const float* h        = (const float*)d_in[1];
  const float* c        = (const float*)d_in[2];
  const float* W_i2h    = (const float*)d_in[3];
  const float* b_i2h    = (const float*)d_in[4];
  const float* W_h2h    = (const float*)d_in[5];
  const float* b_h2h    = (const float*)d_in[6];
  const float* mem_keys = (const float*)d_in[7];
  const float* mem_vals = (const float*)d_in[8];
  const float* W_actor  = (const float*)d_in[9];
  const float* b_actor  = (const float*)d_in[10];
  const float* W_critic = (const float*)d_in[11];
  const float* b_critic = (const float*)d_in[12];

  float* out      = (float*)d_out;
  float* keys_new = out + 273;                            // 16+1+128+128
  float* vals_new = keys_new + (size_t)DICT_LEN * IN_DIM;
  float* ws       = (float*)d_ws;

  k0_preact<<<NBP, 256, 0, stream>>>(x, h, W_i2h, b_i2h, W_h2h, b_h2h, ws);
  k1_keys_pass<<<NB1, 256, 0, stream>>>(x, mem_keys, keys_new, ws);
  k2_reduce_min<<<1, 256, 0, stream>>>(ws);
  k3_exp_pass<<<NB3, 256, 0, stream>>>(ws);
  k4_reduce_sum<<<1, 256, 0, stream>>>(ws);
  k5_vals_pass<<<NB5, 256, 0, stream>>>(mem_vals, vals_new, ws);
  k6_finalize<<<1, 256, 0, stream>>>(x, c, W_actor, b_actor, W_critic, b_critic,
                                     out, keys_new, vals_new, ws);
}